// Generator3DLUT_39822936768697
// MI455X (gfx1250) — compile-verified
//
#include <hip/hip_runtime.h>
#include <stdint.h>

// ---------------------------------------------------------------------------
// 3D LUT (33^3) trilinear apply, (8,3,1024,1024) fp32.
// Roofline: 201 MB stream -> ~8.6 us @ 23.3 TB/s HBM; compute trivial.
// CDNA5 strategy:
//   * LUT repacked fp16 (r,g,b,0) = 287.5 KB -> fits 320 KB WGP LDS
//   * staged once per persistent block via global_load_async_to_lds_b128
//     (ASYNCcnt) + s_wait_asynccnt
//   * 8 ds_load_b64 corner gathers / pixel from LDS
//   * v_fma_mix_f32-friendly accumulation (f16 operand folded into fp32 fma)
//   * nontemporal b128 loads/stores for the read-once/write-once image,
//     global_prefetch_b8 look-ahead; LUT stays hot in cache/LDS
// ---------------------------------------------------------------------------

typedef __attribute__((ext_vector_type(4))) float vf4;

#define LUT_DIM   33
#define N_LUT     (LUT_DIM * LUT_DIM * LUT_DIM)   // 35937
#define N_LUT_PAD (N_LUT + 1)                      // 35938 -> 287,504 B (16B divisible)
#define SMEM_BYTES ((unsigned)(N_LUT_PAD * 8))     // 287,504 <= 320 KB WGP LDS
#define CHUNKS    (SMEM_BYTES / 16u)               // 17,969 x b128 chunks
#define TPB       1024
#define HW4       (1u << 18)                       // float4s per (H*W) plane
#define GROUPS    (8u * HW4)                       // 2,097,152 groups of 4 pixels

struct __align__(8) HEntry { _Float16 r, g, b, pad; };
struct F3 { float r, g, b; };

__device__ __align__(16) HEntry g_lutPacked[N_LUT_PAD];

// ---- Kernel 1: pack fp32 planar LUT -> fp16 interleaved (r,g,b,0) ---------
__global__ void lut_pack_kernel(const float* __restrict__ lut) {
    unsigned e = blockIdx.x * blockDim.x + threadIdx.x;
    if (e >= N_LUT_PAD) return;
    HEntry h;
    if (e < N_LUT) {
        h.r = (_Float16)lut[e];
        h.g = (_Float16)lut[N_LUT + e];
        h.b = (_Float16)lut[2 * N_LUT + e];
    } else {
        h.r = (_Float16)0.0f; h.g = (_Float16)0.0f; h.b = (_Float16)0.0f;
    }
    h.pad = (_Float16)0.0f;
    g_lutPacked[e] = h;
}

// ---- Per-pixel trilinear fetch+blend from LDS -----------------------------
__device__ __forceinline__ F3 apply1(const HEntry* __restrict__ lut_s,
                                     float xr, float xg, float xb) {
    float r = xr * 32.0f, g = xg * 32.0f, b = xb * 32.0f;
    // floor + clamp in float (input >= 0): matches ref ir=clip(floor,0,31), f=r-ir
    float rf = fminf(floorf(r), 31.0f);
    float gf = fminf(floorf(g), 31.0f);
    float bf = fminf(floorf(b), 31.0f);
    float fr = r - rf, fg = g - gf, fb = b - bf;
    int ir = (int)rf, ig = (int)gf, ib = (int)bf;

    int base = ir + ig * LUT_DIM + ib * (LUT_DIM * LUT_DIM);
    HEntry c000 = lut_s[base];
    HEntry c001 = lut_s[base + 1];
    HEntry c010 = lut_s[base + LUT_DIM];
    HEntry c011 = lut_s[base + LUT_DIM + 1];
    HEntry c100 = lut_s[base + LUT_DIM * LUT_DIM];
    HEntry c101 = lut_s[base + LUT_DIM * LUT_DIM + 1];
    HEntry c110 = lut_s[base + LUT_DIM * LUT_DIM + LUT_DIM];
    HEntry c111 = lut_s[base + LUT_DIM * LUT_DIM + LUT_DIM + 1];

    float gr = 1.0f - fr, gg = 1.0f - fg, gb = 1.0f - fb;
    float w00 = gb * gg, w01 = gb * fg, w10 = fb * gg, w11 = fb * fg;
    float w000 = w00 * gr, w001 = w00 * fr, w010 = w01 * gr, w011 = w01 * fr;
    float w100 = w10 * gr, w101 = w10 * fr, w110 = w11 * gr, w111 = w11 * fr;

    // fmaf((float)f16, f32, f32) -> v_fma_mix_f32 (no standalone cvt)
    F3 o;
    float ar_ = (float)c000.r * w000;
    ar_ = fmaf((float)c001.r, w001, ar_);
    ar_ = fmaf((float)c010.r, w010, ar_);
    ar_ = fmaf((float)c011.r, w011, ar_);
    ar_ = fmaf((float)c100.r, w100, ar_);
    ar_ = fmaf((float)c101.r, w101, ar_);
    ar_ = fmaf((float)c110.r, w110, ar_);
    ar_ = fmaf((float)c111.r, w111, ar_);
    o.r = ar_;

    float ag_ = (float)c000.g * w000;
    ag_ = fmaf((float)c001.g, w001, ag_);
    ag_ = fmaf((float)c010.g, w010, ag_);
    ag_ = fmaf((float)c011.g, w011, ag_);
    ag_ = fmaf((float)c100.g, w100, ag_);
    ag_ = fmaf((float)c101.g, w101, ag_);
    ag_ = fmaf((float)c110.g, w110, ag_);
    ag_ = fmaf((float)c111.g, w111, ag_);
    o.g = ag_;

    float ab_ = (float)c000.b * w000;
    ab_ = fmaf((float)c001.b, w001, ab_);
    ab_ = fmaf((float)c010.b, w010, ab_);
    ab_ = fmaf((float)c011.b, w011, ab_);
    ab_ = fmaf((float)c100.b, w100, ab_);
    ab_ = fmaf((float)c101.b, w101, ab_);
    ab_ = fmaf((float)c110.b, w110, ab_);
    ab_ = fmaf((float)c111.b, w111, ab_);
    o.b = ab_;

    return o;
}

// ---- Kernel 2: stage LUT to LDS (async DMA), stream pixels ----------------
__global__ __launch_bounds__(TPB)
void lut_apply_kernel(const float* __restrict__ x, float* __restrict__ out) {
    extern __shared__ char smem_raw[];
    const HEntry* lut_s = reinterpret_cast<const HEntry*>(smem_raw);

    // Stage 287.5 KB packed LUT from global into LDS via CDNA5 async copy.
    // Flat shared-pointer low 32 bits == LDS byte offset (aperture in high 32).
    {
        const uint64_t src     = (uint64_t)(uintptr_t)&g_lutPacked[0];
        const unsigned ldsBase = (unsigned)(uintptr_t)smem_raw;
        for (unsigned i = threadIdx.x; i < CHUNKS; i += TPB) {
            unsigned off = i * 16u;
            unsigned lds = ldsBase + off;
            asm volatile("global_load_async_to_lds_b128 %0, %1, %2"
                         :: "v"(lds), "v"(off), "s"(src) : "memory");
        }
        asm volatile("s_wait_asynccnt 0x0" ::: "memory");
    }
    __syncthreads();

    const vf4* __restrict__ x4 = (const vf4*)x;
    vf4* __restrict__ o4       = (vf4*)out;

    const unsigned stride = (unsigned)gridDim.x * TPB;
    for (unsigned gi = blockIdx.x * TPB + threadIdx.x; gi < GROUPS; gi += stride) {
        unsigned bt = gi >> 18;            // batch index (HW4 = 2^18 float4s)
        unsigned hw = gi & (HW4 - 1u);
        unsigned ar = bt * 3u * HW4 + hw;  // red plane, float4 units

        // Read-once streams: nontemporal so they don't churn cache vs. LUT.
        vf4 vr = __builtin_nontemporal_load(x4 + ar);
        vf4 vg = __builtin_nontemporal_load(x4 + ar + HW4);
        vf4 vb = __builtin_nontemporal_load(x4 + ar + 2u * HW4);

        // Prefetch next iteration's stream addresses.
        unsigned gn = gi + stride;
        if (gn < GROUPS) {
            unsigned btn = gn >> 18, hwn = gn & (HW4 - 1u);
            unsigned arn = btn * 3u * HW4 + hwn;
            __builtin_prefetch(x4 + arn, 0, 1);
            __builtin_prefetch(x4 + arn + HW4, 0, 1);
            __builtin_prefetch(x4 + arn + 2u * HW4, 0, 1);
        }

        F3 p0 = apply1(lut_s, vr.x, vg.x, vb.x);
        F3 p1 = apply1(lut_s, vr.y, vg.y, vb.y);
        F3 p2 = apply1(lut_s, vr.z, vg.z, vb.z);
        F3 p3 = apply1(lut_s, vr.w, vg.w, vb.w);

        vf4 orr, og, ob;
        orr.x = p0.r; orr.y = p1.r; orr.z = p2.r; orr.w = p3.r;
        og.x  = p0.g; og.y  = p1.g; og.z  = p2.g; og.w  = p3.g;
        ob.x  = p0.b; ob.y  = p1.b; ob.z  = p2.b; ob.w  = p3.b;

        // Write-once streams: nontemporal stores.
        __builtin_nontemporal_store(orr, o4 + ar);
        __builtin_nontemporal_store(og,  o4 + ar + HW4);
        __builtin_nontemporal_store(ob,  o4 + ar + 2u * HW4);
    }
}

// ---------------------------------------------------------------------------
extern "C" void kernel_launch(void* const* d_in, const int* in_sizes, int n_in,
                              void* d_out, int out_size, void* d_ws, size_t ws_size,
                              hipStream_t stream) {
    (void)in_sizes; (void)n_in; (void)out_size; (void)d_ws; (void)ws_size;
    const float* lut = (const float*)d_in[0];   // (3, 33,33,33) fp32
    const float* x   = (const float*)d_in[1];   // (8, 3, 1024, 1024) fp32
    float* out       = (float*)d_out;           // (8, 3, 1024, 1024) fp32

    // Persistent grid: one block per compute unit for balanced streaming and
    // a single LUT staging per resident block. Host query is deterministic
    // and graph-capture safe (no stream interaction).
    int blocks = 512;
    int dev = 0;
    if (hipGetDevice(&dev) == hipSuccess) {
        hipDeviceProp_t prop;
        if (hipGetDeviceProperties(&prop, dev) == hipSuccess &&
            prop.multiProcessorCount > 0) {
            blocks = prop.multiProcessorCount;
            if (blocks > 2048) blocks = 2048;
        }
    }

    lut_pack_kernel<<<(N_LUT_PAD + 255) / 256, 256, 0, stream>>>(lut);
    lut_apply_kernel<<<blocks, TPB, SMEM_BYTES, stream>>>(x, out);
}